// CapsuleDense_36215164240457
// MI455X (gfx1250) — compile-verified
//
#include <hip/hip_runtime.h>
#include <cstddef>

// CapsuleDense EM routing, fused recompute design for MI455X (gfx1250).
//   B=32, NI=2048, NO=128, D=16, 3 EM iterations.
// Votes (512 MB) are never materialized: W (16 MB) stays L2-resident and votes
// are recomputed per pass with V_WMMA_F32_16X16X4_F32 (K=4 == the r-contraction,
// N=16 packed as 4 batches x 4 pose columns).

#define B_   32
#define NI_  2048
#define NO_  128
#define D_   16
#define ICH  32            // input capsules per workgroup chunk
#define EPS_ 1e-9f
#define LOG2PI_ 1.8378770664093453f

typedef __attribute__((ext_vector_type(2))) float v2f;
typedef __attribute__((ext_vector_type(8))) float v8f;

union V8U { v8f v; float f[8]; };

// workspace layout (float offsets)
#define OFF_S1   0         // [B*NO*D]  sum Rp*v
#define OFF_S2   65536     // [B*NO*D]  sum Rp*v^2
#define OFF_RS   131072    // [B*NO]    sum Rp
#define OFF_MU   135168    // [B*NO*D]
#define OFF_IS2  200704    // [B*NO*D]  1/sigma2
#define OFF_LS   266240    // [B*NO]    sum_d log sigma2
#define OFF_LOGA 270336    // [B*NO]    log(a_out+eps)

__global__ void em_zero(float* __restrict__ ws) {
  int idx = blockIdx.x * blockDim.x + threadIdx.x;
  if (idx < 2 * 65536 + 4096) ws[idx] = 0.0f;   // S1,S2,Rs are contiguous
}

// One pass over all input capsules. FIRST=1: uniform R=1/NO (iteration 0 stats).
// FIRST=0: recompute votes, logp -> softmax R update, accumulate next stats.
// Block: 256 threads (8 waves). Grid: (NI/ICH, 8 b-groups).
// Wave w handles M-tiles w*4..w*4+3 (4 j's x 4 p's per tile).
template <int FIRST>
__global__ __launch_bounds__(256)
void em_pass(const float* __restrict__ pose, const float* __restrict__ act,
             const float* __restrict__ W,
             const float* __restrict__ muG, const float* __restrict__ is2G,
             const float* __restrict__ lsG, const float* __restrict__ logaG,
             float* __restrict__ S1G, float* __restrict__ S2G,
             float* __restrict__ RsG)
{
  __shared__ float P[512];       // partial sum_d diff^2/sigma2 per (b_local,j)
  __shared__ float red[512];     // reduction scratch
  __shared__ float Rl[512];      // routing weights R per (b_local,j)
  __shared__ float lsl[512];
  __shared__ float logal[512];
  __shared__ float a_lds[4 * ICH];

  const int tid  = threadIdx.x;
  const int wave = tid >> 5;
  const int lane = tid & 31;          // wave32
  const int n    = lane & 15;         // WMMA column: (b_local, q)
  const bool hi  = lane >= 16;
  const int bl   = n >> 2;            // local batch 0..3
  const int q    = n & 3;
  const int k0   = hi ? 2 : 0;        // K pair held by this half-wave

  const int g  = blockIdx.y;          // b-group (4 batches)
  const int i0 = blockIdx.x * ICH;
  const int p0 = tid, p1 = tid + 256; // (b_local,j) pairs: b=p>>7, j=p&127

  {
    if (tid < 4 * ICH) {
      int b = tid / ICH, ii = tid % ICH;
      a_lds[tid] = act[(size_t)(g * 4 + b) * NI_ + (i0 + ii)];
    }
    if (!FIRST) {
      int bj0 = (g * 4 + (p0 >> 7)) * NO_ + (p0 & 127);
      int bj1 = (g * 4 + (p1 >> 7)) * NO_ + (p1 & 127);
      lsl[p0] = lsG[bj0];   lsl[p1] = lsG[bj1];
      logal[p0] = logaG[bj0]; logal[p1] = logaG[bj1];
    }
  }
  __syncthreads();

  float acc1[4][8] = {{0}}, acc2[4][8] = {{0}};
  float rs0 = 0.f, rs1 = 0.f;
  const float invNO = 1.0f / (float)NO_;

  for (int ii = 0; ii < ICH; ++ii) {
    const int i = i0 + ii;

    // B operand: pose[b,i,r,q] packed as N = 4 batches x 4 q columns, K=r.
    const float* pb = pose + (((size_t)(g * 4 + bl) * NI_ + i) << 4) + q;
    v2f bv; bv.x = pb[k0 * 4]; bv.y = pb[k0 * 4 + 4];

    const float* wbase = W + ((size_t)i << 11);   // i * NO*16
    if (ii + 1 < ICH)
      __builtin_prefetch(wbase + 2048 + (tid << 3), 0, 1);  // next W slice

    // votes for 4 M-tiles: C[t] rows = (j,p), cols = (b_local,q)
    V8U C[4];
#pragma unroll
    for (int t = 0; t < 4; ++t) {
      const int mtile = wave * 4 + t;
      const float* pa = wbase + ((mtile * 16 + n) << 2) + k0;  // A: M=n, K pair
      v2f av; av.x = pa[0]; av.y = pa[1];
      v8f cz = {};
      C[t].v = __builtin_amdgcn_wmma_f32_16x16x4_f32(
          false, av, false, bv, (short)0, cz, false, false);
    }

    const float a_in = a_lds[bl * ICH + ii];

    if (FIRST) {
      const float rp = a_in * invNO;
#pragma unroll
      for (int t = 0; t < 4; ++t)
#pragma unroll
        for (int v = 0; v < 8; ++v) {
          float c = C[t].f[v];
          acc1[t][v] += rp * c;
          acc2[t][v] += rp * c * c;
        }
      rs0 += a_lds[(p0 >> 7) * ICH + ii] * invNO;
      rs1 += a_lds[(p1 >> 7) * ICH + ii] * invNO;
    } else {
      P[p0] = 0.f; P[p1] = 0.f;
      __syncthreads();
      // accumulate sum_d (v-mu)^2 / sigma2 into P[b_local][j]
#pragma unroll
      for (int t = 0; t < 4; ++t) {
        const int mtile = wave * 4 + t;
#pragma unroll
        for (int v = 0; v < 8; ++v) {
          int row = mtile * 16 + v + (hi ? 8 : 0);
          int j = row >> 2, d = (row & 3) * 4 + q;
          int idx = ((g * 4 + bl) * NO_ + j) * D_ + d;
          float diff = C[t].f[v] - muG[idx];      // mu/is2 are L2-resident
          atomicAdd(&P[bl * 128 + j], diff * diff * is2G[idx]);
        }
      }
      __syncthreads();
      // logp + log(a_out) -> softmax over j (per local batch)
      float val0 = logal[p0] - 0.5f * (P[p0] + lsl[p0] + 16.f * LOG2PI_);
      float val1 = logal[p1] - 0.5f * (P[p1] + lsl[p1] + 16.f * LOG2PI_);
      red[p0] = val0; red[p1] = val1;
      for (int s = 64; s > 0; s >>= 1) {
        __syncthreads();
        if ((p0 & 127) < s) red[p0] = fmaxf(red[p0], red[p0 + s]);
        if ((p1 & 127) < s) red[p1] = fmaxf(red[p1], red[p1 + s]);
      }
      __syncthreads();
      float m0 = red[p0 & ~127], m1 = red[p1 & ~127];
      float e0 = __expf(val0 - m0), e1 = __expf(val1 - m1);
      __syncthreads();
      red[p0] = e0; red[p1] = e1;
      for (int s = 64; s > 0; s >>= 1) {
        __syncthreads();
        if ((p0 & 127) < s) red[p0] += red[p0 + s];
        if ((p1 & 127) < s) red[p1] += red[p1 + s];
      }
      __syncthreads();
      float r0 = e0 / (red[p0 & ~127] + EPS_);
      float r1 = e1 / (red[p1 & ~127] + EPS_);
      Rl[p0] = r0; Rl[p1] = r1;
      rs0 += r0 * a_lds[(p0 >> 7) * ICH + ii];
      rs1 += r1 * a_lds[(p1 >> 7) * ICH + ii];
      __syncthreads();
      // accumulate next-iteration stats with fresh R (votes still in VGPRs)
#pragma unroll
      for (int t = 0; t < 4; ++t) {
        const int mtile = wave * 4 + t;
#pragma unroll
        for (int v = 0; v < 8; ++v) {
          int row = mtile * 16 + v + (hi ? 8 : 0);
          int j = row >> 2;
          float rp = Rl[bl * 128 + j] * a_in;
          float c = C[t].f[v];
          acc1[t][v] += rp * c;
          acc2[t][v] += rp * c * c;
        }
      }
      __syncthreads();
    }
  }

  // flush register accumulators (each lane owns fixed (b,j,d) slots)
#pragma unroll
  for (int t = 0; t < 4; ++t) {
    const int mtile = wave * 4 + t;
#pragma unroll
    for (int v = 0; v < 8; ++v) {
      int row = mtile * 16 + v + (hi ? 8 : 0);
      int j = row >> 2, d = (row & 3) * 4 + q;
      int idx = ((g * 4 + bl) * NO_ + j) * D_ + d;
      atomicAdd(&S1G[idx], acc1[t][v]);
      atomicAdd(&S2G[idx], acc2[t][v]);
    }
  }
  atomicAdd(&RsG[(g * 4 + (p0 >> 7)) * NO_ + (p0 & 127)], rs0);
  atomicAdd(&RsG[(g * 4 + (p1 >> 7)) * NO_ + (p1 & 127)], rs1);
}

// mu = S1/Rs ; sigma2 = E[(v-mu)^2] + eps via S2 ; a_out ; zero accumulators.
__global__ void em_finalize(const float* __restrict__ beta_v,
                            const float* __restrict__ beta_a,
                            float* __restrict__ S1G, float* __restrict__ S2G,
                            float* __restrict__ RsG,
                            float* __restrict__ muG, float* __restrict__ is2G,
                            float* __restrict__ lsG, float* __restrict__ logaG,
                            float* __restrict__ out, float lam, int final_iter)
{
  int bj = blockIdx.x * blockDim.x + threadIdx.x;
  if (bj >= B_ * NO_) return;
  int j = bj & (NO_ - 1);
  float rsum = RsG[bj];
  float Rs = rsum + EPS_;
  float inv = 1.0f / Rs;
  float ls = 0.f;
#pragma unroll
  for (int d = 0; d < D_; ++d) {
    int idx = bj * D_ + d;
    float s1 = S1G[idx], s2 = S2G[idx];
    float mu = s1 * inv;
    float sig = (s2 - 2.f * mu * s1 + mu * mu * rsum) * inv + EPS_;
    muG[idx] = mu;
    is2G[idx] = 1.0f / sig;
    ls += __logf(sig);
    if (final_iter) out[idx] = mu;       // pose_out = mu.reshape(B,NO,4,4)
    S1G[idx] = 0.f; S2G[idx] = 0.f;      // ready for next accumulation pass
  }
  float cost = (16.f * beta_v[j] + 0.5f * ls) * Rs;
  float a = 1.0f / (1.0f + __expf(-lam * (beta_a[j] - cost)));
  lsG[bj] = ls;
  logaG[bj] = __logf(a + EPS_);
  if (final_iter) out[B_ * NO_ * D_ + bj] = a;
  RsG[bj] = 0.f;
}

extern "C" void kernel_launch(void* const* d_in, const int* in_sizes, int n_in,
                              void* d_out, int out_size, void* d_ws, size_t ws_size,
                              hipStream_t stream) {
  (void)in_sizes; (void)n_in; (void)out_size; (void)ws_size;
  const float* pose = (const float*)d_in[0];
  const float* act  = (const float*)d_in[1];
  const float* W    = (const float*)d_in[2];
  const float* bv   = (const float*)d_in[3];
  const float* ba   = (const float*)d_in[4];
  float* out = (float*)d_out;
  float* ws  = (float*)d_ws;

  float* S1   = ws + OFF_S1;
  float* S2   = ws + OFF_S2;
  float* Rs   = ws + OFF_RS;
  float* mu   = ws + OFF_MU;
  float* is2  = ws + OFF_IS2;
  float* ls   = ws + OFF_LS;
  float* loga = ws + OFF_LOGA;

  const float lam0 = 0.01f * (1.0f - 0.95f);
  const float lam1 = 0.01f * (1.0f - 0.95f * 0.95f);
  const float lam2 = 0.01f * (1.0f - 0.95f * 0.95f * 0.95f);

  dim3 grid(NI_ / ICH, 8);   // 64 i-chunks x 8 batch-groups
  dim3 blk(256);
  dim3 fgrid((B_ * NO_ + 255) / 256);

  em_zero<<<(2 * 65536 + 4096 + 255) / 256, 256, 0, stream>>>(ws);

  // t = 0: uniform routing
  em_pass<1><<<grid, blk, 0, stream>>>(pose, act, W, mu, is2, ls, loga, S1, S2, Rs);
  em_finalize<<<fgrid, 256, 0, stream>>>(bv, ba, S1, S2, Rs, mu, is2, ls, loga, out, lam0, 0);
  // t = 1: R update fused with stats accumulation
  em_pass<0><<<grid, blk, 0, stream>>>(pose, act, W, mu, is2, ls, loga, S1, S2, Rs);
  em_finalize<<<fgrid, 256, 0, stream>>>(bv, ba, S1, S2, Rs, mu, is2, ls, loga, out, lam1, 0);
  // t = 2
  em_pass<0><<<grid, blk, 0, stream>>>(pose, act, W, mu, is2, ls, loga, S1, S2, Rs);
  em_finalize<<<fgrid, 256, 0, stream>>>(bv, ba, S1, S2, Rs, mu, is2, ls, loga, out, lam2, 1);
}